// HybridDiffusion_69715909148720
// MI455X (gfx1250) — compile-verified
//
#include <hip/hip_runtime.h>
#include <cstdint>

#define V_SIZE   32000
#define MASK_IDX 31999
#define SEQ_LEN  1024
// B = 2*exp(-20)/(1-exp(-20))
#define B_CONST  4.1223072533e-09f
#define EPS_C    1.0e-4f

#define BLOCK    256
#define CHUNK_F4 1000                 // 1000 float4 = 4000 floats = 16000 bytes
#define TAIL_F4  (CHUNK_F4 - 3 * BLOCK)   // 232 guarded slots in 4th issue
#define NCHUNK   8                    // 8 * 4000 = 32000 = V
#define CHUNK_BYTES (CHUNK_F4 * 16)

typedef float f4 __attribute__((ext_vector_type(4)));
typedef __attribute__((address_space(3))) float lds_f;

// One lane copies 16 bytes global -> LDS asynchronously (tracked by ASYNCcnt).
__device__ __forceinline__ void async_f4(uint32_t lds_addr, uint32_t goff, const float* base) {
    asm volatile("global_load_async_to_lds_b128 %0, %1, %2"
                 :: "v"(lds_addr), "v"(goff), "s"(base)
                 : "memory");
}
__device__ __forceinline__ void wait_async_le4() {
    asm volatile("s_wait_asynccnt 4" ::: "memory");
}
__device__ __forceinline__ void wait_async_zero() {
    asm volatile("s_wait_asynccnt 0" ::: "memory");
}

// Straight-line: every wave issues exactly 4 async instructions per chunk
// (4th runs with partial EXEC in wave 7 -> still one ASYNCcnt increment).
__device__ __forceinline__ void issue_chunk(const float* u, uint32_t gBase,
                                            uint32_t lBase, uint32_t tid) {
    uint32_t o0 = tid * 16u;
    async_f4(lBase + o0,                 gBase + o0,                 u);
    async_f4(lBase + o0 + 256u * 16u,    gBase + o0 + 256u * 16u,    u);
    async_f4(lBase + o0 + 512u * 16u,    gBase + o0 + 512u * 16u,    u);
    if (tid < TAIL_F4) {
        async_f4(lBase + o0 + 768u * 16u, gBase + o0 + 768u * 16u,   u);
    }
}

__device__ __forceinline__ void scan4(f4 x, int gi, float& bv, int& bi) {
    if (x.x > bv) { bv = x.x; bi = gi;     }
    if (x.y > bv) { bv = x.y; bi = gi + 1; }
    if (x.z > bv) { bv = x.z; bi = gi + 2; }
    if (x.w > bv) { bv = x.w; bi = gi + 3; }
}

__global__ void __launch_bounds__(BLOCK)
hd_sample_kernel(const int* __restrict__ input_ids,
                 const float* __restrict__ tvec,
                 const float* __restrict__ u,
                 float* __restrict__ out) {
    const int row = blockIdx.x;                 // 0 .. BATCH*SEQ-1
    const int tid = threadIdx.x;

    __shared__ __align__(16) float smem[2 * CHUNK_F4 * 4];
    __shared__ float rv[BLOCK / 32];
    __shared__ int   ri[BLOCK / 32];

    const uint32_t smemBase = (uint32_t)(uintptr_t)(lds_f*)smem;   // raw LDS byte offset
    const uint32_t rowByte  = (uint32_t)row * (uint32_t)(V_SIZE * 4);

    // ---- pipelined streaming argmax of u over the row ----
    issue_chunk(u, rowByte, smemBase, (uint32_t)tid);

    float bv = -1.0f;   // u in [0,1): sentinel below all values
    int   bi = 0;

    for (int c = 0; c < NCHUNK; ++c) {
        if (c + 1 < NCHUNK) {
            issue_chunk(u, rowByte + (uint32_t)(c + 1) * CHUNK_BYTES,
                        smemBase + (uint32_t)((c + 1) & 1) * CHUNK_BYTES, (uint32_t)tid);
            wait_async_le4();     // 4 newly issued outstanding => chunk c complete
        } else {
            wait_async_zero();
        }
        __syncthreads();          // publish chunk c to all waves

        const f4* bufp = (const f4*)(smem + (size_t)(c & 1) * (CHUNK_F4 * 4));
        const int base = c * (CHUNK_F4 * 4);

        f4 x0 = bufp[tid];
        f4 x1 = bufp[tid + 256];
        f4 x2 = bufp[tid + 512];
        scan4(x0, base + tid * 4,           bv, bi);
        scan4(x1, base + (tid + 256) * 4,   bv, bi);
        scan4(x2, base + (tid + 512) * 4,   bv, bi);
        if (tid < TAIL_F4) {
            f4 x3 = bufp[tid + 768];
            scan4(x3, base + (tid + 768) * 4, bv, bi);
        }
        __syncthreads();          // all readers done before buffer reuse
    }

    // ---- wave32 reduction (max value, ties -> smaller index) ----
    #pragma unroll
    for (int off = 16; off > 0; off >>= 1) {
        float ov = __shfl_xor(bv, off, 32);
        int   oi = __shfl_xor(bi, off, 32);
        if (ov > bv || (ov == bv && oi < bi)) { bv = ov; bi = oi; }
    }
    if ((tid & 31) == 0) { rv[tid >> 5] = bv; ri[tid >> 5] = bi; }
    __syncthreads();

    if (tid == 0) {
        #pragma unroll
        for (int w = 1; w < BLOCK / 32; ++w) {
            if (rv[w] > bv || (rv[w] == bv && ri[w] < bi)) { bv = rv[w]; bi = ri[w]; }
        }

        // ---- closed-form distribution values for this row ----
        const int   b    = row >> 10;            // SEQ_LEN = 1024
        const float t    = tvec[b];
        const float t1m  = 1.0f - t;
        const float c_t  = sqrtf(t) * sqrtf(t1m) * B_CONST;     // gamma = 1
        const float C    = fmaxf(1.0f + c_t, EPS_C);
        const float alph = t1m / C;
        const float p_bg = (c_t * (1.0f / (float)(V_SIZE - 1))) / C;
        const float p_mk = t / C;

        const int    id   = input_ids[row];
        const float* urow = u + (size_t)row * V_SIZE;
        const float  u_id = urow[id];
        const float  u_mk = urow[MASK_IDX];

        auto prob = [&](int v) -> float {
            float p = (v == MASK_IDX) ? p_mk : p_bg;
            if (v == id) p += alph;
            return p;
        };
        auto score = [&](int v, float uv) -> float {
            float g = 1.0e-10f - logf(uv + 1.0e-10f);   // > 0
            return prob(v) / g;
        };

        // global argmax of probs/g is always among {u-argmax, id, MASK}
        float s0 = score(bi, bv);
        float s1 = score(id, u_id);
        float s2 = score(MASK_IDX, u_mk);

        int   zi = bi; float zs = s0;
        if (s1 > zs || (s1 == zs && id < zi))        { zs = s1; zi = id; }
        if (s2 > zs || (s2 == zs && MASK_IDX < zi))  { zs = s2; zi = MASK_IDX; }

        out[row] = (float)zi;
    }
}

extern "C" void kernel_launch(void* const* d_in, const int* in_sizes, int n_in,
                              void* d_out, int out_size, void* d_ws, size_t ws_size,
                              hipStream_t stream) {
    (void)n_in; (void)d_ws; (void)ws_size; (void)out_size;
    const int*   input_ids = (const int*)d_in[0];
    const float* t         = (const float*)d_in[1];
    const float* u         = (const float*)d_in[2];
    float*       out       = (float*)d_out;

    const int rows = in_sizes[0];   // BATCH * SEQ = 4096
    hd_sample_kernel<<<dim3(rows), dim3(BLOCK), 0, stream>>>(input_ids, t, u, out);
}